// ManifoldDistance_24077586661582
// MI455X (gfx1250) — compile-verified
//
#include <hip/hip_runtime.h>
#include <hip/hip_bf16.h>
#include <math.h>

typedef __attribute__((ext_vector_type(2))) float v2f;
typedef __attribute__((ext_vector_type(8))) float v8f;

#define NCHUNK 4   // column-split waves per 16-row strip (occupancy)

// ---------------------------------------------------------------------------
// Kernel 1: global stats. scalars[0] = s = 1/(max||coord||+1e-6)
//           scalars[1] = eps0 = ||max(pts)-min(pts)||^2  (diameter^2, P=2)
//           scalars[2] = eps_final = BLUR^P = 1e-4
// ---------------------------------------------------------------------------
__global__ __launch_bounds__(1024) void sk_stats(const float* __restrict__ x,
                                                 const float* __restrict__ y,
                                                 int nx, int ny,
                                                 float* __restrict__ scalars)
{
    int t = threadIdx.x;
    float maxn2 = 0.0f;
    float mn[6], mx[6];
#pragma unroll
    for (int d = 0; d < 6; ++d) { mn[d] = 3.0e38f; mx[d] = -3.0e38f; }

    int total = nx + ny;
    for (int p = t; p < total; p += 1024) {
        const float* pt = (p < nx) ? (x + (size_t)p * 6) : (y + (size_t)(p - nx) * 6);
        float v[6];
#pragma unroll
        for (int d = 0; d < 6; ++d) v[d] = pt[d];
        float n2 = v[0]*v[0] + v[1]*v[1] + v[2]*v[2];
        maxn2 = fmaxf(maxn2, n2);
#pragma unroll
        for (int d = 0; d < 6; ++d) { mn[d] = fminf(mn[d], v[d]); mx[d] = fmaxf(mx[d], v[d]); }
    }

#pragma unroll
    for (int off = 16; off >= 1; off >>= 1) {
        maxn2 = fmaxf(maxn2, __shfl_xor(maxn2, off, 32));
#pragma unroll
        for (int d = 0; d < 6; ++d) {
            mn[d] = fminf(mn[d], __shfl_xor(mn[d], off, 32));
            mx[d] = fmaxf(mx[d], __shfl_xor(mx[d], off, 32));
        }
    }

    __shared__ float sm[32][13];
    int wid = t >> 5, lid = t & 31;
    if (lid == 0) {
        sm[wid][0] = maxn2;
#pragma unroll
        for (int d = 0; d < 6; ++d) { sm[wid][1 + d] = mn[d]; sm[wid][7 + d] = mx[d]; }
    }
    __syncthreads();
    if (wid == 0) {
        maxn2 = sm[lid][0];
#pragma unroll
        for (int d = 0; d < 6; ++d) { mn[d] = sm[lid][1 + d]; mx[d] = sm[lid][7 + d]; }
#pragma unroll
        for (int off = 16; off >= 1; off >>= 1) {
            maxn2 = fmaxf(maxn2, __shfl_xor(maxn2, off, 32));
#pragma unroll
            for (int d = 0; d < 6; ++d) {
                mn[d] = fminf(mn[d], __shfl_xor(mn[d], off, 32));
                mx[d] = fmaxf(mx[d], __shfl_xor(mx[d], off, 32));
            }
        }
        if (lid == 0) {
            scalars[0] = 1.0f / (sqrtf(maxn2) + 1e-6f);
            float d2 = 0.0f;
#pragma unroll
            for (int d = 0; d < 6; ++d) { float df = mx[d] - mn[d]; d2 += df * df; }
            scalars[1] = d2;       // eps0 = diameter^2
            scalars[2] = 1.0e-4f;  // eps_final
        }
    }
}

// ---------------------------------------------------------------------------
// Kernel 2: pack points, fragment-permuted so each lane's WMMA operand pair is
// one contiguous 16B load:  mem layout per point = [K0,K1,K4,K5, K2,K3,0,0].
// K = [s*coord(3), 0.1*clamp(col,0,1)(3)]   (0.1 = sqrt(LAMBDA_COLOR))
// ---------------------------------------------------------------------------
__global__ void sk_pack(const float* __restrict__ src,
                        const float* __restrict__ scalars,
                        float* __restrict__ dst8,
                        float* __restrict__ sqn, int npts)
{
    int i = blockIdx.x * blockDim.x + threadIdx.x;
    if (i >= npts) return;
    float s = scalars[0];
    const float* p = src + (size_t)i * 6;
    float c[6];
    c[0] = p[0] * s; c[1] = p[1] * s; c[2] = p[2] * s;
    c[3] = 0.1f * fminf(fmaxf(p[3], 0.0f), 1.0f);
    c[4] = 0.1f * fminf(fmaxf(p[4], 0.0f), 1.0f);
    c[5] = 0.1f * fminf(fmaxf(p[5], 0.0f), 1.0f);
    float sq = 0.0f;
#pragma unroll
    for (int k = 0; k < 6; ++k) sq += c[k] * c[k];
    float* d = dst8 + (size_t)i * 8;
    d[0] = c[0]; d[1] = c[1]; d[2] = c[4]; d[3] = c[5];   // half-wave 0: K0,1 | K4,5
    d[4] = c[2]; d[5] = c[3]; d[6] = 0.0f; d[7] = 0.0f;   // half-wave 1: K2,3 | K6,7
    sqn[i] = sq;
}

// ---------------------------------------------------------------------------
// Kernel 3: partial softmin.  grid = strips * NCHUNK waves; each wave owns a
// 16-row strip and 1/NCHUNK of the columns.  Per 16-col tile:
//   G = P(16x8) * Q(8x16) via 2x V_WMMA_F32_16X16X4_F32 (fp32 A/B: required
//   because C/eps_final scales errors by 1e4 -> f16 Gram would be garbage),
//   cost = max(sq_r + sq_c - 2G, 0), online (max,sum) logsumexp of h_j-cost/eps.
// Emits per-row partial (max,sum) pairs; sk_combine merges them.
// ---------------------------------------------------------------------------
template <bool USEH>
__global__ __launch_bounds__(32) void sk_softmin_part(
    const float* __restrict__ Prow, const float* __restrict__ sqRow,
    const float* __restrict__ Qcol, const float* __restrict__ sqCol,
    const float* __restrict__ hvec,   // used iff USEH: extra h term = hvec/eps
    float2* __restrict__ part,        // [total_rows, NCHUNK] (m, s) pairs
    const float* __restrict__ epsPtr,
    float hbase, int Nrow, int Ncol)
{
    const int strips_per_batch = Nrow >> 4;
    const int strip = blockIdx.x / NCHUNK;
    const int chunk = blockIdx.x % NCHUNK;
    const int b  = strip / strips_per_batch;
    const int r0 = (strip % strips_per_batch) << 4;

    const int lane = threadIdx.x;
    const int hi = lane >> 4;      // half-wave: K pair (0,1) vs (2,3)
    const int ln = lane & 15;      // row (A) / col (B) index within tile

    const float inv_eps = 1.0f / epsPtr[0];

    const size_t rowbase = (size_t)b * Nrow;
    const size_t colbase = (size_t)b * Ncol;

    // A fragments: one contiguous b128 per lane thanks to permuted packing.
    const float4 af = *(const float4*)(Prow + (rowbase + r0 + ln) * 8 + 4 * hi);
    v2f a0, a1;
    a0[0] = af.x; a0[1] = af.y;    // K slice 0..3 operand
    a1[0] = af.z; a1[1] = af.w;    // K slice 4..7 operand

    // C/D layout: element (vgpr v, lane l) = row (v + 8*(l/16)), col (l%16)
    float sqr[8];
#pragma unroll
    for (int v = 0; v < 8; ++v) sqr[v] = sqRow[rowbase + r0 + v + 8 * hi];

    float rm[8], rs[8];
#pragma unroll
    for (int v = 0; v < 8; ++v) { rm[v] = -__builtin_inff(); rs[v] = 0.0f; }

    const int cols_per = Ncol / NCHUNK;
    const int j_begin = chunk * cols_per;
    const int j_end   = j_begin + cols_per;

    for (int j0 = j_begin; j0 < j_end; j0 += 16) {
        const int col = j0 + ln;
        const float* qr = Qcol + (colbase + col) * 8;
        const float4 bf = *(const float4*)(qr + 4 * hi);
        // speculative, high-locality prefetch of next tile (WGP-scope path)
        __builtin_prefetch((const void*)(qr + 16 * 8), 0, 3);
        v2f b0, b1;
        b0[0] = bf.x; b0[1] = bf.y;
        b1[0] = bf.z; b1[1] = bf.w;

        v8f acc = {0.f, 0.f, 0.f, 0.f, 0.f, 0.f, 0.f, 0.f};
        acc = __builtin_amdgcn_wmma_f32_16x16x4_f32(false, a0, false, b0,
                                                    (short)0, acc, false, false);
        acc = __builtin_amdgcn_wmma_f32_16x16x4_f32(false, a1, false, b1,
                                                    (short)0, acc, false, false);

        const float sqc = sqCol[colbase + col];
        float hj = hbase;
        if (USEH) hj += hvec[colbase + col] * inv_eps;

#pragma unroll
        for (int v = 0; v < 8; ++v) {
            float cost = fmaxf(sqr[v] + sqc - 2.0f * acc[v], 0.0f);
            float tt = hj - cost * inv_eps;
            float nm = fmaxf(rm[v], tt);
            rs[v] = rs[v] * __expf(rm[v] - nm) + __expf(tt - nm);
            rm[v] = nm;
        }
    }

    // combine partial (max,sum) across the 16 lanes holding each row
#pragma unroll
    for (int off = 1; off < 16; off <<= 1) {
#pragma unroll
        for (int v = 0; v < 8; ++v) {
            float om = __shfl_xor(rm[v], off, 32);
            float os = __shfl_xor(rs[v], off, 32);
            float nm = fmaxf(rm[v], om);
            rs[v] = rs[v] * __expf(rm[v] - nm) + os * __expf(om - nm);
            rm[v] = nm;
        }
    }

    if (ln == 0) {
#pragma unroll
        for (int v = 0; v < 8; ++v) {
            int row = r0 + v + 8 * hi;
            part[(rowbase + row) * NCHUNK + chunk] = make_float2(rm[v], rs[v]);
        }
    }
}

// ---------------------------------------------------------------------------
// Kernel 3b: merge NCHUNK partial (m,s) pairs per row, finish the softmin,
// optionally average with previous iterate (simultaneous f,g update).
// ---------------------------------------------------------------------------
__global__ void sk_combine(const float2* __restrict__ part,
                           const float* __restrict__ mixOld,  // nullable
                           float* __restrict__ out,
                           const float* __restrict__ epsPtr, int total)
{
    int g = blockIdx.x * blockDim.x + threadIdx.x;
    if (g >= total) return;
    const float eps = epsPtr[0];
    float m = -__builtin_inff(), s = 0.0f;
#pragma unroll
    for (int c = 0; c < NCHUNK; ++c) {
        float2 p = part[(size_t)g * NCHUNK + c];
        float nm = fmaxf(m, p.x);
        s = s * __expf(m - nm) + p.y * __expf(p.x - nm);
        m = nm;
    }
    float val = -eps * (m + __logf(s));
    if (mixOld) val = 0.5f * (mixOld[g] + val);
    out[g] = val;
}

// ---------------------------------------------------------------------------
// Kernel 4: out[0] = 10 * mean_b( mean_i f + mean_j g )
// ---------------------------------------------------------------------------
__global__ __launch_bounds__(1024) void sk_finalize(const float* __restrict__ fn,
                                                    const float* __restrict__ gn,
                                                    float* __restrict__ out,
                                                    int nf, int ng,
                                                    float scale_f, float scale_g)
{
    int t = threadIdx.x;
    float sf = 0.0f, sg = 0.0f;
    for (int i = t; i < nf; i += 1024) sf += fn[i];
    for (int i = t; i < ng; i += 1024) sg += gn[i];
    float s = scale_f * sf + scale_g * sg;
#pragma unroll
    for (int off = 16; off >= 1; off >>= 1) s += __shfl_xor(s, off, 32);
    __shared__ float sm[32];
    int wid = t >> 5, lid = t & 31;
    if (lid == 0) sm[wid] = s;
    __syncthreads();
    if (wid == 0) {
        s = sm[lid];
#pragma unroll
        for (int off = 16; off >= 1; off >>= 1) s += __shfl_xor(s, off, 32);
        if (lid == 0) out[0] = s;
    }
}

// ---------------------------------------------------------------------------
static inline void softmin_pass(const float* P, const float* sqP,
                                const float* Q, const float* sqQ,
                                const float* hvec, const float* mixOld,
                                float2* part, float* out, const float* epsPtr,
                                float hbase, int Nrow, int Ncol, int B,
                                hipStream_t stream)
{
    const int totalRows = B * Nrow;
    const int grid = (totalRows / 16) * NCHUNK;
    if (hvec)
        sk_softmin_part<true><<<grid, 32, 0, stream>>>(P, sqP, Q, sqQ, hvec, part,
                                                       epsPtr, hbase, Nrow, Ncol);
    else
        sk_softmin_part<false><<<grid, 32, 0, stream>>>(P, sqP, Q, sqQ, nullptr, part,
                                                        epsPtr, hbase, Nrow, Ncol);
    sk_combine<<<(totalRows + 255) / 256, 256, 0, stream>>>(part, mixOld, out,
                                                            epsPtr, totalRows);
}

extern "C" void kernel_launch(void* const* d_in, const int* in_sizes, int n_in,
                              void* d_out, int out_size, void* d_ws, size_t ws_size,
                              hipStream_t stream)
{
    const float* x = (const float*)d_in[0];
    const float* y = (const float*)d_in[1];
    float* outp = (float*)d_out;

    const int B = 4;
    const int N = in_sizes[0] / (B * 6);   // 4096
    const int M = in_sizes[1] / (B * 6);   // 4096
    const int NP = B * N;
    const int MP = B * M;
    const int PMAX = (NP > MP) ? NP : MP;

    // workspace carve-up
    char* w = (char*)d_ws;
    float*  scalars = (float*)w;           w += 256;
    float*  Px  = (float*)w;               w += (size_t)NP * 8 * sizeof(float);
    float*  Py  = (float*)w;               w += (size_t)MP * 8 * sizeof(float);
    float*  sqx = (float*)w;               w += (size_t)NP * sizeof(float);
    float*  sqy = (float*)w;               w += (size_t)MP * sizeof(float);
    float2* part = (float2*)w;             w += (size_t)PMAX * NCHUNK * sizeof(float2);
    float*  f0  = (float*)w;               w += (size_t)NP * sizeof(float);
    float*  f1  = (float*)w;               w += (size_t)NP * sizeof(float);
    float*  f2  = (float*)w;               w += (size_t)NP * sizeof(float);
    float*  fn  = (float*)w;               w += (size_t)NP * sizeof(float);
    float*  g0  = (float*)w;               w += (size_t)MP * sizeof(float);
    float*  g1  = (float*)w;               w += (size_t)MP * sizeof(float);
    float*  g2  = (float*)w;               w += (size_t)MP * sizeof(float);
    float*  gn  = (float*)w;               w += (size_t)MP * sizeof(float);

    const float* eps0 = scalars + 1;
    const float* epsf = scalars + 2;
    const float hb_f = -logf((float)M);   // b_log constant
    const float hb_g = -logf((float)N);   // a_log constant

    sk_stats<<<1, 1024, 0, stream>>>(x, y, NP, MP, scalars);
    sk_pack<<<(NP + 255) / 256, 256, 0, stream>>>(x, scalars, Px, sqx, NP);
    sk_pack<<<(MP + 255) / 256, 256, 0, stream>>>(y, scalars, Py, sqy, MP);

    // f0 = softmin(eps0, C_xy, b);  g0 = softmin(eps0, C_yx, a)
    softmin_pass(Px, sqx, Py, sqy, nullptr, nullptr, part, f0, eps0, hb_f, N, M, B, stream);
    softmin_pass(Py, sqy, Px, sqx, nullptr, nullptr, part, g0, eps0, hb_g, M, N, B, stream);
    // iter 1 (eps0), simultaneous update with 0.5 averaging
    softmin_pass(Px, sqx, Py, sqy, g0, f0, part, f1, eps0, hb_f, N, M, B, stream);
    softmin_pass(Py, sqy, Px, sqx, f0, g0, part, g1, eps0, hb_g, M, N, B, stream);
    // iter 2 (eps_final)
    softmin_pass(Px, sqx, Py, sqy, g1, f1, part, f2, epsf, hb_f, N, M, B, stream);
    softmin_pass(Py, sqy, Px, sqx, f1, g1, part, g2, epsf, hb_g, M, N, B, stream);
    // final pass (no averaging)
    softmin_pass(Px, sqx, Py, sqy, g2, nullptr, part, fn, epsf, hb_f, N, M, B, stream);
    softmin_pass(Py, sqy, Px, sqx, f2, nullptr, part, gn, epsf, hb_g, M, N, B, stream);

    sk_finalize<<<1, 1024, 0, stream>>>(fn, gn, outp, NP, MP, 10.0f / NP, 10.0f / MP);
}